// EGNN_19078244729030
// MI455X (gfx1250) — compile-verified
//
#include <hip/hip_runtime.h>

typedef __attribute__((ext_vector_type(2))) float v2f;
typedef __attribute__((ext_vector_type(8))) float v8f;

#define DF 16
#define DH 64
#define DM 32
#define NN 1024
#define BB 4
#define ROWS (BB * NN)   // 4096

// ---------------------------------------------------------------------------
// fp32 WMMA tile: D(16x16) += A(16x4) * B(4x16), per CDNA5 VGPR layouts.
// A: lane L holds M=L&15, K = kbase + {0,1} (lanes 0-15) / kbase + {2,3} (16-31)
// B: lane L holds N=L&15, same K split.
// C/D: VGPR r -> M = r (lanes 0-15) / 8+r (lanes 16-31), N = lane&15.
// ---------------------------------------------------------------------------

__device__ __forceinline__ float fast_tanh(float x) {
    float e2 = __expf(2.0f * x);
    return 1.0f - 2.0f / (e2 + 1.0f);   // correct limits at +/-inf
}

// h0 = node_feat @ W_in + b_in   [4096x16]@[16x64]
__global__ void k_embed(const float* __restrict__ X, const float* __restrict__ W,
                        const float* __restrict__ bias, float* __restrict__ H) {
    int lane = threadIdx.x & 31;
    int gw = blockIdx.x * (blockDim.x >> 5) + (threadIdx.x >> 5);
    int m0 = (gw >> 2) * 16;
    int n0 = (gw & 3) * 16;
    int mi = m0 + (lane & 15);
    int ni = n0 + (lane & 15);
    int ko = (lane >> 4) << 1;
    v8f acc = {};
#pragma unroll
    for (int k = 0; k < DF; k += 4) {
        v2f a, b;
        a.x = X[mi * DF + k + ko];
        a.y = X[mi * DF + k + ko + 1];
        b.x = W[(k + ko) * DH + ni];
        b.y = W[(k + ko + 1) * DH + ni];
        acc = __builtin_amdgcn_wmma_f32_16x16x4_f32(false, a, false, b,
                                                    (short)0, acc, false, false);
    }
    int half = lane >> 4;
#pragma unroll
    for (int r = 0; r < 8; r++) {
        int row = m0 + r + half * 8;
        H[row * DH + ni] = acc[r] + bias[ni];
    }
}

// pre = h@Wh1 + bh1 + bd + bx + x@Wx ; mjb = h@Wh2 + bh2    (per layer l)
__global__ void k_proj(const float* __restrict__ H,
                       const float* __restrict__ Wh1, const float* __restrict__ bh1,
                       const float* __restrict__ Wh2, const float* __restrict__ bh2,
                       const float* __restrict__ bd,
                       const float* __restrict__ Wx, const float* __restrict__ bx,
                       const float* __restrict__ pos,
                       float* __restrict__ pre, float* __restrict__ mjb, int l) {
    int lane = threadIdx.x & 31;
    int gw = blockIdx.x * (blockDim.x >> 5) + (threadIdx.x >> 5);
    int m_tile = gw >> 2, q = gw & 3;
    int sel = q >> 1;            // 0: Wh1 -> pre,  1: Wh2 -> mjb
    int n0 = (q & 1) * 16;
    int m0 = m_tile * 16;
    const float* W = (sel ? Wh2 : Wh1) + l * DH * DM;
    int mi = m0 + (lane & 15);
    int ni = n0 + (lane & 15);
    int ko = (lane >> 4) << 1;
    v8f acc = {};
#pragma unroll
    for (int k = 0; k < DH; k += 4) {
        v2f a, b;
        a.x = H[mi * DH + k + ko];
        a.y = H[mi * DH + k + ko + 1];
        b.x = W[(k + ko) * DM + ni];
        b.y = W[(k + ko + 1) * DM + ni];
        acc = __builtin_amdgcn_wmma_f32_16x16x4_f32(false, a, false, b,
                                                    (short)0, acc, false, false);
    }
    int half = lane >> 4;
    if (sel == 0) {
        float bsum = bh1[l * DM + ni] + bd[l * DM + ni] + bx[l * DM + ni];
        float wx0 = Wx[l * 3 * DM + 0 * DM + ni];
        float wx1 = Wx[l * 3 * DM + 1 * DM + ni];
        float wx2 = Wx[l * 3 * DM + 2 * DM + ni];
#pragma unroll
        for (int r = 0; r < 8; r++) {
            int row = m0 + r + half * 8;
            float x0 = pos[row * 3 + 0], x1 = pos[row * 3 + 1], x2 = pos[row * 3 + 2];
            pre[row * DM + ni] = acc[r] + bsum + x0 * wx0 + x1 * wx1 + x2 * wx2;
        }
    } else {
        float b2 = bh2[l * DM + ni];
#pragma unroll
        for (int r = 0; r < 8; r++) {
            int row = m0 + r + half * 8;
            mjb[row * DM + ni] = acc[r] + b2;
        }
    }
}

// Heavy pairwise kernel: one wave per node i, lanes over j (32 at a time),
// 32-channel message loop unrolled in registers.
__global__ void __launch_bounds__(256) k_pair(
        const float* __restrict__ pre, const float* __restrict__ mjb,
        const float* __restrict__ pos, const int* __restrict__ edges,
        const float* __restrict__ Wd, const float* __restrict__ Wg,
        const float* __restrict__ bg,
        float* __restrict__ agg, float* __restrict__ v, int l) {
    int lane = threadIdx.x & 31;
    int w = threadIdx.x >> 5;
    int ig = blockIdx.x * 8 + w;       // global row (b*N + i)
    int b = ig >> 10;

    float Ac[DM], Wdc[DM], Wgc[DM], accv[DM];
    const float* prow = pre + ig * DM;
#pragma unroll
    for (int c = 0; c < DM; c++) {
        Ac[c]  = prow[c];
        Wdc[c] = Wd[l * DM + c];
        Wgc[c] = Wg[l * DM + c];
        accv[c] = 0.f;
    }
    float bgl = bg[l];
    float xi0 = pos[ig * 3 + 0], xi1 = pos[ig * 3 + 1], xi2 = pos[ig * 3 + 2];
    const int*   erow = edges + (size_t)ig * NN;
    const float* posb = pos + b * NN * 3;
    const float* mjbb = mjb + b * NN * DM;
    float cs0 = 0.f, cs1 = 0.f, cs2 = 0.f;

    for (int jt = 0; jt < NN; jt += 32) {
        int j = jt + lane;
        float e = (float)erow[j];
        float xj0 = posb[j * 3 + 0], xj1 = posb[j * 3 + 1], xj2 = posb[j * 3 + 2];
        float r0 = xi0 - xj0, r1 = xi1 - xj1, r2 = xi2 - xj2;
        float dsq = r0 * r0 + r1 * r1 + r2 * r2;
        const float4* mrow = (const float4*)(mjbb + j * DM);
        float mj[DM];
#pragma unroll
        for (int q = 0; q < 8; q++) {
            float4 t = mrow[q];
            mj[q * 4 + 0] = t.x; mj[q * 4 + 1] = t.y;
            mj[q * 4 + 2] = t.z; mj[q * 4 + 3] = t.w;
        }
        float gacc = 0.f;
#pragma unroll
        for (int c = 0; c < DM; c++) {
            float m = fmaf(dsq, Wdc[c], Ac[c]) + mj[c];
            m = fmaxf(m, 0.f) * e;              // masked message
            accv[c] += m;
            gacc = fmaf(m, Wgc[c], gacc);
        }
        float ge = fast_tanh(gacc + bgl) * e;
        cs0 = fmaf(r0, ge, cs0);
        cs1 = fmaf(r1, ge, cs1);
        cs2 = fmaf(r2, ge, cs2);
    }

    // cross-lane (j) reductions, wave32 butterflies
    float outv = 0.f;
#pragma unroll
    for (int c = 0; c < DM; c++) {
        float s = accv[c];
        for (int off = 16; off > 0; off >>= 1) s += __shfl_xor(s, off, 32);
        if (lane == c) outv = s;
    }
    agg[ig * DM + lane] = outv;

    for (int off = 16; off > 0; off >>= 1) {
        cs0 += __shfl_xor(cs0, off, 32);
        cs1 += __shfl_xor(cs1, off, 32);
        cs2 += __shfl_xor(cs2, off, 32);
    }
    if (lane < 3) {
        float add = lane == 0 ? cs0 : (lane == 1 ? cs1 : cs2);
        v[ig * 3 + lane] += add;
    }
}

// h_next = h + relu(concat(h, agg) @ Wu + bu)   [4096x96]@[96x64]
__global__ void k_update(const float* __restrict__ Hin, const float* __restrict__ agg,
                         const float* __restrict__ Wu, const float* __restrict__ bu,
                         float* __restrict__ Hout, int l) {
    int lane = threadIdx.x & 31;
    int gw = blockIdx.x * (blockDim.x >> 5) + (threadIdx.x >> 5);
    int m0 = (gw >> 2) * 16;
    int n0 = (gw & 3) * 16;
    const float* W = Wu + l * (DH + DM) * DH;
    int mi = m0 + (lane & 15);
    int ni = n0 + (lane & 15);
    int ko = (lane >> 4) << 1;
    v8f acc = {};
#pragma unroll
    for (int k = 0; k < DH + DM; k += 4) {
        v2f a, b;
        int kk = k + ko;
        if (k < DH) {
            a.x = Hin[mi * DH + kk];
            a.y = Hin[mi * DH + kk + 1];
        } else {
            a.x = agg[mi * DM + kk - DH];
            a.y = agg[mi * DM + kk - DH + 1];
        }
        b.x = W[kk * DH + ni];
        b.y = W[(kk + 1) * DH + ni];
        acc = __builtin_amdgcn_wmma_f32_16x16x4_f32(false, a, false, b,
                                                    (short)0, acc, false, false);
    }
    int half = lane >> 4;
    float bun = bu[l * DH + ni];
#pragma unroll
    for (int r = 0; r < 8; r++) {
        int row = m0 + r + half * 8;
        float val = fmaxf(acc[r] + bun, 0.f);
        Hout[row * DH + ni] = Hin[row * DH + ni] + val;
    }
}

__global__ void k_final(const float* __restrict__ pos, const float* __restrict__ v,
                        float* __restrict__ out) {
    int idx = blockIdx.x * blockDim.x + threadIdx.x;
    if (idx < ROWS * 3) out[idx] = pos[idx] + v[idx];
}

extern "C" void kernel_launch(void* const* d_in, const int* in_sizes, int n_in,
                              void* d_out, int out_size, void* d_ws, size_t ws_size,
                              hipStream_t stream) {
    const float* node_feat = (const float*)d_in[0];
    const float* pos  = (const float*)d_in[1];
    const float* vel  = (const float*)d_in[2];
    const int*   edges = (const int*)d_in[3];
    const float* W_in = (const float*)d_in[4];
    const float* b_in = (const float*)d_in[5];
    const float* Wh1  = (const float*)d_in[6];
    const float* bh1  = (const float*)d_in[7];
    const float* Wh2  = (const float*)d_in[8];
    const float* bh2  = (const float*)d_in[9];
    const float* Wd   = (const float*)d_in[10];
    const float* bd   = (const float*)d_in[11];
    const float* Wx   = (const float*)d_in[12];
    const float* bx   = (const float*)d_in[13];
    const float* Wg   = (const float*)d_in[14];
    const float* bg   = (const float*)d_in[15];
    const float* Wu   = (const float*)d_in[16];
    const float* bu   = (const float*)d_in[17];

    float* ws  = (float*)d_ws;
    float* hA  = ws;                    // 4096*64
    float* hB  = hA + ROWS * DH;        // 4096*64
    float* pre = hB + ROWS * DH;        // 4096*32
    float* mjb = pre + ROWS * DM;       // 4096*32
    float* agg = mjb + ROWS * DM;       // 4096*32
    float* v   = agg + ROWS * DM;       // 4096*3

    hipMemcpyAsync(v, vel, (size_t)ROWS * 3 * sizeof(float),
                   hipMemcpyDeviceToDevice, stream);

    dim3 blk(128);
    k_embed<<<256, blk, 0, stream>>>(node_feat, W_in, b_in, hA);

    float* hc = hA;
    float* hn = hB;
    for (int l = 0; l < 4; l++) {
        k_proj<<<256, blk, 0, stream>>>(hc, Wh1, bh1, Wh2, bh2, bd, Wx, bx,
                                        pos, pre, mjb, l);
        k_pair<<<512, dim3(256), 0, stream>>>(pre, mjb, pos, edges, Wd, Wg, bg,
                                              agg, v, l);
        k_update<<<256, blk, 0, stream>>>(hc, agg, Wu, bu, hn, l);
        float* t = hc; hc = hn; hn = t;
    }
    k_final<<<48, dim3(256), 0, stream>>>(pos, v, (float*)d_out);
}